// Model_29772713296145
// MI455X (gfx1250) — compile-verified
//
#include <hip/hip_runtime.h>
#include <math.h>

// Problem constants (from reference)
#define BB   8
#define LL   336
#define HH   192
#define CC   321
#define KK   16
#define SS   8
#define NH2  97            // H/2+1
#define OUTF 3201          // NH2*(2K+1)
#define SEQL 528           // L+H
#define BC   (BB*CC)       // 2568
#define AKD  136           // 128 + 8  (A-row width for filt GEMM)
#define NF   194           // 2*NH2 (re|im columns of DFT output)
#define MIXK 582           // 2*3*NH2 (re|im of mix_in)
#define MIXKP 584          // padded to multiple of 4

typedef __attribute__((ext_vector_type(2))) float v2f;
typedef __attribute__((ext_vector_type(8))) float v8f;

#define TWO_PI 6.28318530717958647692f

// B matrices are stored PACKED for WMMA fragment loads:
//   Bp viewed as v2f[K/2][N]; Bp[kp][n] = (B[2kp][n], B[2kp+1][n])
// so a lane's (k, k+1) B pair is one aligned 8-byte load.
#define PACKED_IDX(row, col, ncols) (((row) >> 1) * (2 * (ncols)) + 2 * (col) + ((row) & 1))

// ---------------------------------------------------------------------------
// 1) DFT matrix F[t][n] (packed), t=K dim (192), n: [0,97)=cos, [97,194)=-sin
// ---------------------------------------------------------------------------
__global__ void k_precompute_F(float* __restrict__ F) {
    int idx = blockIdx.x * blockDim.x + threadIdx.x;
    if (idx >= HH * NF) return;
    int t = idx / NF, n = idx % NF;
    float v;
    if (n < NH2) {
        int ph = (n * t) % HH;
        v = cosf(TWO_PI * (float)ph / (float)HH);
    } else {
        int m = n - NH2;
        int ph = (m * t) % HH;
        v = -sinf(TWO_PI * (float)ph / (float)HH);
    }
    F[PACKED_IDX(t, n, NF)] = v;
}

// ---------------------------------------------------------------------------
// 2) BG[j][t] = (Bmix @ G)[j][t]  (fused complex-mix + irfft), packed (584,192)
//    rows 582..583 zero; plus ybias[t] (irfft of mix_b, plain vector).
// ---------------------------------------------------------------------------
__global__ void k_precompute_BG(const float* __restrict__ wr,   // (97,291)
                                const float* __restrict__ wi,   // (97,291)
                                const float* __restrict__ br,   // (97,)
                                const float* __restrict__ bi,   // (97,)
                                float* __restrict__ BG,         // packed (584,192)
                                float* __restrict__ ybias) {    // (192,)
    int idx = blockIdx.x * blockDim.x + threadIdx.x;
    if (idx >= (MIXKP + 1) * HH) return;
    int j = idx / HH, t = idx % HH;
    if (j >= MIXK && j < MIXKP) { BG[PACKED_IDX(j, t, HH)] = 0.f; return; }
    float acc = 0.f;
    for (int h = 0; h < NH2; ++h) {
        int ph = (h * t) % HH;
        float ang = TWO_PI * (float)ph / (float)HH;
        float whf = (h == 0 || h == NH2 - 1) ? 1.f : 2.f;
        float Gr = whf * cosf(ang) / (float)HH;
        float Gi = -whf * sinf(ang) / (float)HH;
        if (j < 291) {
            acc += wr[h * 291 + j] * Gr + wi[h * 291 + j] * Gi;
        } else if (j < MIXK) {
            int j2 = j - 291;
            acc += -wi[h * 291 + j2] * Gr + wr[h * 291 + j2] * Gi;
        } else { // j == MIXKP: bias row
            acc += br[h] * Gr + bi[h] * Gi;
        }
    }
    if (j < MIXK) BG[PACKED_IDX(j, t, HH)] = acc;
    else          ybias[t] = acc;
}

// ---------------------------------------------------------------------------
// 2b) Wcat packed (136,3201) = [mh_w (128,3201); mh_b (8,3201)]
// ---------------------------------------------------------------------------
__global__ void k_build_wcat(const float* __restrict__ mh_w,
                             const float* __restrict__ mh_b,
                             float* __restrict__ Wcat) {
    int idx = blockIdx.x * blockDim.x + threadIdx.x;
    if (idx >= AKD * OUTF) return;
    int rI = idx / OUTF, col = idx % OUTF;
    float v = (rI < 128) ? mh_w[idx] : mh_b[idx - 128 * OUTF];
    Wcat[PACKED_IDX(rI, col, OUTF)] = v;
}

// ---------------------------------------------------------------------------
// 3) Per-(b,c): instance norm, normalized seq, classifier softmax p,
//    correlation softmax cf, A-row [p[s]*cf[k] (128) | p[s] (8)].
// ---------------------------------------------------------------------------
__global__ void k_norm_p(const float* __restrict__ x,        // (B,L,C)
                         const float* __restrict__ yh,       // (B,H,C)
                         const float* __restrict__ rr,       // (B,C,K)
                         const float* __restrict__ temp,     // (1,)
                         const float* __restrict__ cls_w,    // (S,L)
                         const float* __restrict__ cls_b,    // (S,)
                         const float* __restrict__ bstate,   // (C,S)
                         float* __restrict__ seq,            // (BC,528)
                         float* __restrict__ muA,            // (BC,)
                         float* __restrict__ stdA,           // (BC,)
                         float* __restrict__ Arows) {        // (BC,136)
    __shared__ float red[128];
    __shared__ float res[10];   // sum, sumsq, dot[0..7]
    int bc = blockIdx.x;
    int b = bc / CC, c = bc % CC;
    int tid = threadIdx.x;
    const float* xb = x + b * LL * CC + c;   // stride CC
    const float* yb = yh + b * HH * CC + c;  // stride CC

    float s = 0.f, s2 = 0.f;
    float dp[SS];
#pragma unroll
    for (int q = 0; q < SS; ++q) dp[q] = 0.f;
    for (int l = tid; l < LL; l += 128) {
        float v = xb[l * CC];
        s += v; s2 += v * v;
#pragma unroll
        for (int q = 0; q < SS; ++q) dp[q] += v * cls_w[q * LL + l];
    }
    float vals[10];
    vals[0] = s; vals[1] = s2;
#pragma unroll
    for (int q = 0; q < SS; ++q) vals[2 + q] = dp[q];
    for (int q = 0; q < 10; ++q) {
        red[tid] = vals[q]; __syncthreads();
        for (int off = 64; off > 0; off >>= 1) {
            if (tid < off) red[tid] += red[tid + off];
            __syncthreads();
        }
        if (tid == 0) res[q] = red[0];
        __syncthreads();
    }
    float mu = res[0] / (float)LL;
    float var = res[1] / (float)LL - mu * mu;
    float sd = sqrtf(var + 1e-8f);
    float inv = 1.f / sd;

    float* srow = seq + bc * SEQL;
    for (int l = tid; l < LL; l += 128) srow[l] = (xb[l * CC] - mu) * inv;
    for (int t = tid; t < HH; t += 128) srow[LL + t] = (yb[t * CC] - mu) * inv;

    if (tid == 0) {
        muA[bc] = mu; stdA[bc] = sd;
        float lg[SS], mx = -1e30f;
#pragma unroll
        for (int q = 0; q < SS; ++q) {
            lg[q] = cls_b[q] + bstate[c * SS + q] + res[2 + q];
            if (lg[q] > mx) mx = lg[q];
        }
        float den = 0.f;
#pragma unroll
        for (int q = 0; q < SS; ++q) { lg[q] = expf(lg[q] - mx); den += lg[q]; }
        float p[SS];
#pragma unroll
        for (int q = 0; q < SS; ++q) p[q] = lg[q] / den;
        float T = temp[0];
        float a0 = 1.f / T, m2 = a0;
        float av[KK];
#pragma unroll
        for (int k = 0; k < KK; ++k) {
            av[k] = fabsf(rr[bc * KK + k]) / T;
            if (av[k] > m2) m2 = av[k];
        }
        float den2 = expf(a0 - m2);
#pragma unroll
        for (int k = 0; k < KK; ++k) { av[k] = expf(av[k] - m2); den2 += av[k]; }
        float* arow = Arows + bc * AKD;
#pragma unroll
        for (int q = 0; q < SS; ++q) {
#pragma unroll
            for (int k = 0; k < KK; ++k) arow[q * KK + k] = p[q] * (av[k] / den2);
            arow[128 + q] = p[q];
        }
    }
}

// ---------------------------------------------------------------------------
// Generic f32 WMMA GEMM: C(M,N) = A(M,K) @ B(K,N).
// A row-major; B PACKED (v2f[K/2][N]); one wave per 32x64 tile:
// 8 WMMAs per K-step from 2 A-pair loads + 4 B-pair loads (all b64).
// Branchless inner loop (clamped loads, guarded stores). Kd%4==0, lda even.
// ---------------------------------------------------------------------------
__global__ void k_gemm_f32(const float* __restrict__ A, int lda,
                           const float* __restrict__ Bp, int N,
                           float* __restrict__ Cp, int ldc,
                           int M, int Kd) {
    int lane = threadIdx.x;
    int lo = lane & 15, hi = lane >> 4;
    int m32 = blockIdx.y * 32;
    int n64 = blockIdx.x * 64;
    int ar0 = m32 + lo;      if (ar0 > M - 1) ar0 = M - 1;     // clamp
    int ar1 = m32 + 16 + lo; if (ar1 > M - 1) ar1 = M - 1;     // clamp
    const v2f* ap0 = (const v2f*)(A + ar0 * lda) + hi;
    const v2f* ap1 = (const v2f*)(A + ar1 * lda) + hi;
    const v2f* bp  = (const v2f*)Bp + hi * N;
    int bn[4];
#pragma unroll
    for (int j = 0; j < 4; ++j) {
        int n = n64 + j * 16 + lo;
        bn[j] = (n > N - 1) ? (N - 1) : n;                     // clamp
    }
    v8f acc00 = {}, acc01 = {}, acc02 = {}, acc03 = {};
    v8f acc10 = {}, acc11 = {}, acc12 = {}, acc13 = {};
    for (int k0 = 0; k0 < Kd; k0 += 4) {
        int kh = k0 >> 1;
        v2f a0 = ap0[kh];
        v2f a1 = ap1[kh];
        const v2f* Brow = bp + kh * N;
        v2f b0 = Brow[bn[0]], b1 = Brow[bn[1]], b2 = Brow[bn[2]], b3 = Brow[bn[3]];
        acc00 = __builtin_amdgcn_wmma_f32_16x16x4_f32(false, a0, false, b0, (short)0, acc00, false, false);
        acc01 = __builtin_amdgcn_wmma_f32_16x16x4_f32(false, a0, false, b1, (short)0, acc01, false, false);
        acc02 = __builtin_amdgcn_wmma_f32_16x16x4_f32(false, a0, false, b2, (short)0, acc02, false, false);
        acc03 = __builtin_amdgcn_wmma_f32_16x16x4_f32(false, a0, false, b3, (short)0, acc03, false, false);
        acc10 = __builtin_amdgcn_wmma_f32_16x16x4_f32(false, a1, false, b0, (short)0, acc10, false, false);
        acc11 = __builtin_amdgcn_wmma_f32_16x16x4_f32(false, a1, false, b1, (short)0, acc11, false, false);
        acc12 = __builtin_amdgcn_wmma_f32_16x16x4_f32(false, a1, false, b2, (short)0, acc12, false, false);
        acc13 = __builtin_amdgcn_wmma_f32_16x16x4_f32(false, a1, false, b3, (short)0, acc13, false, false);
    }
    v8f accs[2][4] = {{acc00, acc01, acc02, acc03}, {acc10, acc11, acc12, acc13}};
#pragma unroll
    for (int half = 0; half < 2; ++half) {
#pragma unroll
        for (int j = 0; j < 4; ++j) {
            int n = n64 + j * 16 + lo;
            if (n >= N) continue;
#pragma unroll
            for (int rI = 0; rI < 8; ++rI) {
                int m = m32 + half * 16 + rI + 8 * hi;
                if (m < M) Cp[m * ldc + n] = accs[half][j][rI];
            }
        }
    }
}

// ---------------------------------------------------------------------------
// sf GEMM with fused gather: rows = (bc*16 + k), A[row][t] =
//   sign(r[bc,k]) * seq[b*C + leader[bc,k]][L + t - shift[bc,k]]
// B = packed F. M = 41088 (multiple of 32). 32x64 tiles.
// ---------------------------------------------------------------------------
__global__ void k_sf_gemm(const float* __restrict__ seq,
                          const int* __restrict__ leader,   // (B,C,K)
                          const int* __restrict__ shift,    // (B,C,K)
                          const float* __restrict__ rr,     // (B,C,K)
                          const float* __restrict__ Fp,     // packed (192,194)
                          float* __restrict__ sf) {         // (BC*16,194)
    int lane = threadIdx.x;
    int lo = lane & 15, hi = lane >> 4;
    int m32 = blockIdx.y * 32;
    int n64 = blockIdx.x * 64;
    int row0 = m32 + lo;
    int row1 = row0 + 16;
    int bc0 = row0 >> 4, bc1 = row1 >> 4;
    int ld0 = leader[row0], ld1 = leader[row1];     // (bc*16+k) == flat r-index
    int sh0 = shift[row0],  sh1 = shift[row1];
    float rv0 = rr[row0], rv1 = rr[row1];
    float sg0 = (rv0 > 0.f) ? 1.f : ((rv0 < 0.f) ? -1.f : 0.f);
    float sg1 = (rv1 > 0.f) ? 1.f : ((rv1 < 0.f) ? -1.f : 0.f);
    const float* srow0 = seq + ((bc0 / CC) * CC + ld0) * SEQL + (LL - sh0) + 2 * hi;
    const float* srow1 = seq + ((bc1 / CC) * CC + ld1) * SEQL + (LL - sh1) + 2 * hi;
    const v2f* bp = (const v2f*)Fp + hi * NF;
    int bn[4];
#pragma unroll
    for (int j = 0; j < 4; ++j) {
        int n = n64 + j * 16 + lo;
        bn[j] = (n > NF - 1) ? (NF - 1) : n;
    }
    v8f acc00 = {}, acc01 = {}, acc02 = {}, acc03 = {};
    v8f acc10 = {}, acc11 = {}, acc12 = {}, acc13 = {};
    for (int k0 = 0; k0 < HH; k0 += 4) {
        v2f a0, a1;
        a0.x = sg0 * srow0[k0]; a0.y = sg0 * srow0[k0 + 1];
        a1.x = sg1 * srow1[k0]; a1.y = sg1 * srow1[k0 + 1];
        const v2f* Brow = bp + (k0 >> 1) * NF;
        v2f b0 = Brow[bn[0]], b1 = Brow[bn[1]], b2 = Brow[bn[2]], b3 = Brow[bn[3]];
        acc00 = __builtin_amdgcn_wmma_f32_16x16x4_f32(false, a0, false, b0, (short)0, acc00, false, false);
        acc01 = __builtin_amdgcn_wmma_f32_16x16x4_f32(false, a0, false, b1, (short)0, acc01, false, false);
        acc02 = __builtin_amdgcn_wmma_f32_16x16x4_f32(false, a0, false, b2, (short)0, acc02, false, false);
        acc03 = __builtin_amdgcn_wmma_f32_16x16x4_f32(false, a0, false, b3, (short)0, acc03, false, false);
        acc10 = __builtin_amdgcn_wmma_f32_16x16x4_f32(false, a1, false, b0, (short)0, acc10, false, false);
        acc11 = __builtin_amdgcn_wmma_f32_16x16x4_f32(false, a1, false, b1, (short)0, acc11, false, false);
        acc12 = __builtin_amdgcn_wmma_f32_16x16x4_f32(false, a1, false, b2, (short)0, acc12, false, false);
        acc13 = __builtin_amdgcn_wmma_f32_16x16x4_f32(false, a1, false, b3, (short)0, acc13, false, false);
    }
    v8f accs[2][4] = {{acc00, acc01, acc02, acc03}, {acc10, acc11, acc12, acc13}};
#pragma unroll
    for (int half = 0; half < 2; ++half) {
#pragma unroll
        for (int j = 0; j < 4; ++j) {
            int n = n64 + j * 16 + lo;
            if (n >= NF) continue;
#pragma unroll
            for (int rI = 0; rI < 8; ++rI) {
                int m = m32 + half * 16 + rI + 8 * hi;
                sf[m * NF + n] = accs[half][j][rI];
            }
        }
    }
}

// ---------------------------------------------------------------------------
// Combine: apply filt, reduce over k, build mix_in rows (pad cols zeroed)
// ---------------------------------------------------------------------------
__global__ void k_combine(const float* __restrict__ sf,    // (BC*16,194)
                          const float* __restrict__ yf,    // (BC,194)
                          const float* __restrict__ filt,  // (BC,3201)
                          float* __restrict__ mixin) {     // (BC,584)
    int idx = blockIdx.x * blockDim.x + threadIdx.x;
    if (idx >= BC * NH2) return;
    int bc = idx / NH2, h = idx % NH2;
    const float* fb = filt + bc * OUTF;
    const float* sb = sf + bc * KK * NF;
    float yfr = yf[bc * NF + h];
    float yfi = yf[bc * NF + NH2 + h];
    float srs = 0.f, sis = 0.f, drs = 0.f, dis = 0.f;
#pragma unroll 4
    for (int k = 0; k < KK; ++k) {
        float f1 = fb[k * NH2 + h];
        float f2 = fb[(KK + k) * NH2 + h];
        float ar = sb[k * NF + h] * f1;
        float ai = sb[k * NF + NH2 + h] * f1;
        srs += ar; sis += ai;
        drs += (ar - yfr) * f2;
        dis += (ai - yfi) * f2;
    }
    float f3 = fb[2 * KK * NH2 + h];
    float* mi = mixin + bc * MIXKP;
    mi[h]            = srs;
    mi[NH2 + h]      = drs;
    mi[2 * NH2 + h]  = yfr * f3;
    mi[291 + h]           = sis;
    mi[291 + NH2 + h]     = dis;
    mi[291 + 2 * NH2 + h] = yfi * f3;
    if (h == 0) { mi[MIXK] = 0.f; mi[MIXK + 1] = 0.f; }   // K padding
}

// ---------------------------------------------------------------------------
// Final: out[b,t,c] = (yhn + ymix + ybias[t]) * std + mu   (write-coalesced)
// ---------------------------------------------------------------------------
__global__ void k_final(const float* __restrict__ seq,
                        const float* __restrict__ ymix,   // (BC,192)
                        const float* __restrict__ ybias,  // (192,)
                        const float* __restrict__ muA,
                        const float* __restrict__ stdA,
                        float* __restrict__ out) {        // (B,H,C)
    int idx = blockIdx.x * blockDim.x + threadIdx.x;
    if (idx >= BB * HH * CC) return;
    int c = idx % CC;
    int bt = idx / CC;
    int t = bt % HH, b = bt / HH;
    int bc = b * CC + c;
    float y = seq[bc * SEQL + LL + t] + ymix[bc * HH + t] + ybias[t];
    out[idx] = y * stdA[bc] + muA[bc];
}

// ---------------------------------------------------------------------------
extern "C" void kernel_launch(void* const* d_in, const int* in_sizes, int n_in,
                              void* d_out, int out_size, void* d_ws, size_t ws_size,
                              hipStream_t stream) {
    const float* x      = (const float*)d_in[0];
    const float* y_hat  = (const float*)d_in[1];
    const int*   leader = (const int*)d_in[2];
    const int*   shift  = (const int*)d_in[3];
    const float* r      = (const float*)d_in[4];
    const float* temp   = (const float*)d_in[5];
    const float* cls_w  = (const float*)d_in[6];
    const float* cls_b  = (const float*)d_in[7];
    const float* bstate = (const float*)d_in[8];
    const float* mh_w   = (const float*)d_in[9];   // (8, 16*3201) == (128,3201)
    const float* mh_b   = (const float*)d_in[10];  // (8, 3201)
    const float* mwr    = (const float*)d_in[11];
    const float* mwi    = (const float*)d_in[12];
    const float* mbr    = (const float*)d_in[13];
    const float* mbi    = (const float*)d_in[14];
    float* out = (float*)d_out;

    float* wsf = (float*)d_ws;
    size_t off = 0;
    float* Fp    = wsf + off; off += (size_t)HH * NF;        // packed 192*194
    float* BGp   = wsf + off; off += (size_t)MIXKP * HH;     // packed 584*192
    float* ybias = wsf + off; off += HH;
    float* seq   = wsf + off; off += (size_t)BC * SEQL;
    float* muA   = wsf + off; off += BC;
    float* stdA  = wsf + off; off += BC;
    float* Arows = wsf + off; off += (size_t)BC * AKD;
    float* Wcat  = wsf + off; off += (size_t)AKD * OUTF;     // packed 136*3201
    float* filt  = wsf + off; off += (size_t)BC * OUTF;
    float* sf    = wsf + off; off += (size_t)BC * KK * NF;
    float* yf    = wsf + off; off += (size_t)BC * NF;
    float* mixin = wsf + off; off += (size_t)BC * MIXKP;
    float* ymix  = wsf + off; off += (size_t)BC * HH;

    // 1) twiddle / fused-matrix precompute + Wcat (all written pre-packed)
    {
        int n = HH * NF;
        k_precompute_F<<<(n + 255) / 256, 256, 0, stream>>>(Fp);
    }
    {
        int n = (MIXKP + 1) * HH;
        k_precompute_BG<<<(n + 255) / 256, 256, 0, stream>>>(mwr, mwi, mbr, mbi, BGp, ybias);
    }
    {
        int n = AKD * OUTF;
        k_build_wcat<<<(n + 255) / 256, 256, 0, stream>>>(mh_w, mh_b, Wcat);
    }
    // 2) norm / softmaxes / A-rows
    k_norm_p<<<BC, 128, 0, stream>>>(x, y_hat, r, temp, cls_w, cls_b, bstate,
                                     seq, muA, stdA, Arows);
    // 3) filt = [q|p] @ Wcat   (single WMMA GEMM, K=136)
    {
        dim3 grid((OUTF + 63) / 64, (BC + 31) / 32);
        k_gemm_f32<<<grid, 32, 0, stream>>>(Arows, AKD, Wcat, OUTF, filt, OUTF,
                                            BC, AKD);
    }
    // 4) sf = gathered/shifted seq @ F  (fused gather, WMMA DFT)
    {
        dim3 grid((NF + 63) / 64, (BC * KK) / 32);
        k_sf_gemm<<<grid, 32, 0, stream>>>(seq, leader, shift, r, Fp, sf);
    }
    // 5) yf = yhn @ F  (WMMA DFT)
    {
        dim3 grid((NF + 63) / 64, (BC + 31) / 32);
        k_gemm_f32<<<grid, 32, 0, stream>>>(seq + LL, SEQL, Fp, NF, yf, NF,
                                            BC, HH);
    }
    // 6) combine -> mix_in
    {
        int n = BC * NH2;
        k_combine<<<(n + 255) / 256, 256, 0, stream>>>(sf, yf, filt, mixin);
    }
    // 7) ymix = mix_in @ BG  (fused complex mix + irfft, WMMA, K padded to 584)
    {
        dim3 grid((HH + 63) / 64, (BC + 31) / 32);
        k_gemm_f32<<<grid, 32, 0, stream>>>(mixin, MIXKP, BGp, HH, ymix, HH,
                                            BC, MIXKP);
    }
    // 8) final denorm + transpose
    {
        int n = BB * HH * CC;
        k_final<<<(n + 255) / 256, 256, 0, stream>>>(seq, ymix, ybias, muA, stdA, out);
    }
}